// SimpleGIN_57672820850887
// MI455X (gfx1250) — compile-verified
//
#include <hip/hip_runtime.h>
#include <hip/hip_bf16.h>
#include <math.h>

#define NN     50000
#define NE     800000
#define NF     9
#define NVOCAB 100
#define HID    64
#define NG     1024

typedef __attribute__((ext_vector_type(2))) float v2f;
typedef __attribute__((ext_vector_type(8))) float v8f;

// ---------------- AtomEncoder: h[n,c] = sum_f emb[f, x[n,f], c] ----------------
__global__ void gin_encode(const int* __restrict__ x,
                           const float* __restrict__ emb,
                           float* __restrict__ h) {
  int t = blockIdx.x * blockDim.x + threadIdx.x;
  if (t >= NN * HID) return;
  int n = t >> 6, c = t & 63;
  float acc = 0.f;
#pragma unroll
  for (int f = 0; f < NF; ++f) {
    int idx = x[n * NF + f];
    acc += emb[(f * NVOCAB + idx) * HID + c];
  }
  h[t] = acc;
}

// ---------------- z = h (float4 copy) ----------------
__global__ void gin_copy(float4* __restrict__ dst, const float4* __restrict__ src, int n4) {
  int t = blockIdx.x * blockDim.x + threadIdx.x;
  if (t < n4) dst[t] = src[t];
}

// ---------------- z[dst] += h[src], one thread per (edge, channel) ----------------
__global__ void gin_scatter(float* __restrict__ z, const float* __restrict__ h,
                            const int* __restrict__ src, const int* __restrict__ dst) {
  int t = blockIdx.x * blockDim.x + threadIdx.x;
  if (t >= NE * HID) return;
  int e = t >> 6, c = t & 63;
  atomicAdd(&z[dst[e] * HID + c], h[src[e] * HID + c]);
}

// ---------------- fused 2-layer MLP: hout = relu(relu(z@w1+b1)@w2+b2) ----------------
// Block = 128 threads = 4 waves; block processes 16 nodes; wave w owns output cols [16w,16w+16).
// NN is an exact multiple of 16 (3125 blocks), so EXEC is all-ones for every WMMA.
__global__ __launch_bounds__(128)
void gin_mlp(const float* __restrict__ z,
             const float* __restrict__ w1, const float* __restrict__ b1,
             const float* __restrict__ w2, const float* __restrict__ b2,
             float* __restrict__ hout) {
  __shared__ float w1s[HID * 65];   // stride 65 kills LDS bank conflicts
  __shared__ float w2s[HID * 65];
  __shared__ float zt[16 * 65];
  __shared__ float tt[16 * 65];

  const int tid   = threadIdx.x;
  const int node0 = blockIdx.x * 16;

  for (int i = tid; i < HID * HID; i += 128) {
    int r = i >> 6, c = i & 63;
    w1s[r * 65 + c] = w1[i];
    w2s[r * 65 + c] = w2[i];
  }
  for (int i = tid; i < 16 * HID; i += 128) {
    int r = i >> 6, c = i & 63;
    zt[r * 65 + c] = z[(node0 + r) * HID + c];
  }
  __syncthreads();

  const int lane = tid & 31;       // wave32
  const int wave = tid >> 5;
  const int m    = lane & 15;      // A row / B col / D col
  const int hi   = lane >> 4;      // K half-select and D row-half
  const int c0   = wave * 16;

  // ---- GEMM1: t = relu(z @ w1 + b1) ----
  v8f acc = {};
#pragma unroll
  for (int k0 = 0; k0 < HID; k0 += 4) {
    const int kk = k0 + 2 * hi;    // ISA 16x4 f32 A layout: K = vgpr + 2*(lane/16)
    v2f a, b;
    a.x = zt[m * 65 + kk];
    a.y = zt[m * 65 + kk + 1];
    b.x = w1s[kk * 65 + c0 + m];
    b.y = w1s[(kk + 1) * 65 + c0 + m];
    acc = __builtin_amdgcn_wmma_f32_16x16x4_f32(false, a, false, b,
                                                (short)0, acc, false, false);
  }
  const float bias1 = b1[c0 + m];
#pragma unroll
  for (int r = 0; r < 8; ++r) {
    float v = acc[r] + bias1;                 // D: row = r + 8*hi, col = c0 + m
    tt[(r + 8 * hi) * 65 + c0 + m] = v > 0.f ? v : 0.f;
  }
  __syncthreads();

  // ---- GEMM2: hout = relu(t @ w2 + b2) ----
  v8f acc2 = {};
#pragma unroll
  for (int k0 = 0; k0 < HID; k0 += 4) {
    const int kk = k0 + 2 * hi;
    v2f a, b;
    a.x = tt[m * 65 + kk];
    a.y = tt[m * 65 + kk + 1];
    b.x = w2s[kk * 65 + c0 + m];
    b.y = w2s[(kk + 1) * 65 + c0 + m];
    acc2 = __builtin_amdgcn_wmma_f32_16x16x4_f32(false, a, false, b,
                                                 (short)0, acc2, false, false);
  }
  const float bias2 = b2[c0 + m];
#pragma unroll
  for (int r = 0; r < 8; ++r) {
    float v = acc2[r] + bias2;
    hout[(node0 + r + 8 * hi) * HID + c0 + m] = v > 0.f ? v : 0.f;
  }
}

// ---------------- readout ----------------
__global__ void gin_zero(float* __restrict__ p, int n) {
  int t = blockIdx.x * blockDim.x + threadIdx.x;
  if (t < n) p[t] = 0.f;
}

__global__ void gin_pool(const float* __restrict__ h, const int* __restrict__ batch,
                         float* __restrict__ sums, float* __restrict__ counts) {
  int t = blockIdx.x * blockDim.x + threadIdx.x;
  if (t >= NN * HID) return;
  int n = t >> 6, c = t & 63;
  int g = batch[n];
  atomicAdd(&sums[g * HID + c], h[t]);
  if (c == 0) atomicAdd(&counts[g], 1.0f);
}

__global__ void gin_head(const float* __restrict__ sums, const float* __restrict__ counts,
                         const float* __restrict__ lw, const float* __restrict__ lb,
                         float* __restrict__ out) {
  int g = blockIdx.x * blockDim.x + threadIdx.x;
  if (g >= NG) return;
  float cnt = counts[g];
  cnt = cnt > 1.f ? cnt : 1.f;
  float acc = lb[0];
#pragma unroll
  for (int c = 0; c < HID; ++c)
    acc += (sums[g * HID + c] / cnt) * lw[c];
  out[g] = 1.f / (1.f + expf(-acc));
}

extern "C" void kernel_launch(void* const* d_in, const int* in_sizes, int n_in,
                              void* d_out, int out_size, void* d_ws, size_t ws_size,
                              hipStream_t stream) {
  const int*   x        = (const int*)d_in[0];
  const int*   eidx     = (const int*)d_in[1];   // [2, NE] flat
  const int*   batch    = (const int*)d_in[2];
  const float* atom_emb = (const float*)d_in[3];
  const float* cw1[3] = {(const float*)d_in[4],  (const float*)d_in[8],  (const float*)d_in[12]};
  const float* cb1[3] = {(const float*)d_in[5],  (const float*)d_in[9],  (const float*)d_in[13]};
  const float* cw2[3] = {(const float*)d_in[6],  (const float*)d_in[10], (const float*)d_in[14]};
  const float* cb2[3] = {(const float*)d_in[7],  (const float*)d_in[11], (const float*)d_in[15]};
  const float* lin_w = (const float*)d_in[16];
  const float* lin_b = (const float*)d_in[17];
  float* out = (float*)d_out;

  float* ws     = (float*)d_ws;
  float* h      = ws;                        // NN*HID
  float* z      = ws + (size_t)NN * HID;     // NN*HID
  float* sums   = z  + (size_t)NN * HID;     // NG*HID
  float* counts = sums + (size_t)NG * HID;   // NG

  const int* src = eidx;
  const int* dst = eidx + NE;

  // encode
  gin_encode<<<(NN * HID + 255) / 256, 256, 0, stream>>>(x, atom_emb, h);

  // three GIN convs (mlp writes back over h)
  for (int l = 0; l < 3; ++l) {
    gin_copy<<<(NN * HID / 4 + 255) / 256, 256, 0, stream>>>(
        (float4*)z, (const float4*)h, NN * HID / 4);
    gin_scatter<<<(NE * HID + 255) / 256, 256, 0, stream>>>(z, h, src, dst);
    gin_mlp<<<NN / 16, 128, 0, stream>>>(z, cw1[l], cb1[l], cw2[l], cb2[l], h);
  }

  // mean readout + sigmoid head
  gin_zero<<<(NG * HID + NG + 255) / 256, 256, 0, stream>>>(sums, NG * HID + NG);
  gin_pool<<<(NN * HID + 255) / 256, 256, 0, stream>>>(h, batch, sums, counts);
  gin_head<<<(NG + 255) / 256, 256, 0, stream>>>(sums, counts, lin_w, lin_b, out);
}